// Task_Query_Loss_44495861186849
// MI455X (gfx1250) — compile-verified
//
#include <hip/hip_runtime.h>

#define GRID_N   100
#define T_LEN    2048
#define B_LEN    4096
#define ROW_VEC4 (T_LEN * 2 / 4)                     // 1024 float4 per batch row
#define NVEC4    ((long)B_LEN * (long)T_LEN * 2 / 4) // 4,194,304 float4 per tensor
#define THREADS  256
#define BLOCKS_P1 1024
#define INV_N    (1.0f / 16777216.0f)                // 1 / (4096*2048*2), exact pow2

typedef __attribute__((ext_vector_type(2))) float v2f;
typedef __attribute__((ext_vector_type(8))) float v8f;

// Reduce one wave's 32 per-lane accumulators to a single scalar using the
// matrix pipe: D = A(16x4) x ones(4x16). With A[m][0]=acc(lane m),
// A[m][2]=acc(lane m+16), every column of D holds the 16 pairwise sums;
// summing a lane's 8 D VGPRs + one shfl_xor(16) gives the full wave sum.
__device__ __forceinline__ float wave_sum_wmma(float acc) {
    v2f a; a.x = acc;  a.y = 0.0f;
    v2f b; b.x = 1.0f; b.y = 1.0f;
    v8f c = {};
    c = __builtin_amdgcn_wmma_f32_16x16x4_f32(
            /*neg_a=*/false, a, /*neg_b=*/false, b,
            /*c_mod=*/(short)0, c, /*reuse_a=*/false, /*reuse_b=*/false);
    float s = c[0] + c[1] + c[2] + c[3] + c[4] + c[5] + c[6] + c[7];
    s += __shfl_xor(s, 16, 32);
    return s;
}

__global__ __launch_bounds__(THREADS) void loss_partial_kernel(
    const float* __restrict__ traj1, const float* __restrict__ traj2,
    const float* __restrict__ grid_table, float* __restrict__ partials) {
    __shared__ float s_w2[T_LEN];           // w[t]^2 (sqrt cancels vs square)
    __shared__ float s_wave[THREADS / 32];

    // Build w^2 table from trajectory2[0, t, :] + grid_table gather.
    // Source data (16KB + 40KB) is L2-hot; per-block recompute is cheap.
    for (int t = (int)threadIdx.x; t < T_LEN; t += THREADS) {
        float lat = traj2[2 * t + 0];
        float lon = traj2[2 * t + 1];
        int lg = (int)floorf((lat + 1.0f) * 0.5f * (float)GRID_N);
        int og = (int)floorf((lon + 1.0f) * 0.5f * (float)GRID_N);
        lg = min(max(lg, 0), GRID_N - 1);
        og = min(max(og, 0), GRID_N - 1);
        s_w2[t] = grid_table[lg * GRID_N + og] * 1.0e-4f + 1.0f;
    }
    __syncthreads();

    // Stream both tensors as float4 (global_load_b128). Each vec4 covers two
    // consecutive timesteps (t0 even) and never crosses a batch row.
    const float4* a4 = (const float4*)traj1;
    const float4* b4 = (const float4*)traj2;
    float acc = 0.0f;
    const long stride = (long)gridDim.x * THREADS;
    for (long v = (long)blockIdx.x * THREADS + threadIdx.x; v < NVEC4; v += stride) {
        float4 a = a4[v];
        float4 b = b4[v];
        int t0 = ((int)(v & (long)(ROW_VEC4 - 1))) * 2;   // 2*(v % 1024)
        float2 w = *(const float2*)(s_w2 + t0);           // conflict-free b64
        float dx = a.x - b.x, dy = a.y - b.y;
        float dz = a.z - b.z, dw = a.w - b.w;
        acc += w.x * (dx * dx + dy * dy) + w.y * (dz * dz + dw * dw);
    }

    // Wave reduce via WMMA, then cross-wave via LDS.
    float ws = wave_sum_wmma(acc);
    int wave = (int)threadIdx.x >> 5;
    if ((threadIdx.x & 31) == 0) s_wave[wave] = ws;
    __syncthreads();
    if (threadIdx.x == 0) {
        float bsum = 0.0f;
#pragma unroll
        for (int i = 0; i < THREADS / 32; ++i) bsum += s_wave[i];
        partials[blockIdx.x] = bsum;
    }
}

__global__ __launch_bounds__(THREADS) void loss_final_kernel(
    const float* __restrict__ partials, int n, float* __restrict__ out) {
    __shared__ float s_wave[THREADS / 32];
    float acc = 0.0f;
    for (int i = (int)threadIdx.x; i < n; i += THREADS) acc += partials[i];
    float ws = wave_sum_wmma(acc);
    int wave = (int)threadIdx.x >> 5;
    if ((threadIdx.x & 31) == 0) s_wave[wave] = ws;
    __syncthreads();
    if (threadIdx.x == 0) {
        float total = 0.0f;
#pragma unroll
        for (int i = 0; i < THREADS / 32; ++i) total += s_wave[i];
        out[0] = total * INV_N;
    }
}

extern "C" void kernel_launch(void* const* d_in, const int* in_sizes, int n_in,
                              void* d_out, int out_size, void* d_ws, size_t ws_size,
                              hipStream_t stream) {
    (void)in_sizes; (void)n_in; (void)out_size; (void)ws_size;
    const float* traj1      = (const float*)d_in[0];   // (4096, 2048, 2) f32
    const float* traj2      = (const float*)d_in[1];   // (4096, 2048, 2) f32
    const float* grid_table = (const float*)d_in[2];   // (100, 100) f32
    float* out      = (float*)d_out;
    float* partials = (float*)d_ws;                    // 1024 floats of scratch

    loss_partial_kernel<<<BLOCKS_P1, THREADS, 0, stream>>>(traj1, traj2, grid_table, partials);
    loss_final_kernel<<<1, THREADS, 0, stream>>>(partials, BLOCKS_P1, out);
}